// WeightSharedSAS_87608742904000
// MI455X (gfx1250) — compile-verified
//
#include <hip/hip_runtime.h>
#include <math.h>

#define HID 128
#define PAD 132   // LDS row pitch in floats (avoids 64-bank conflicts on stride-128)

typedef float v2f __attribute__((ext_vector_type(2)));
typedef float v8f __attribute__((ext_vector_type(8)));

// ---------------------------------------------------------------------------
// Edge prep: int64 -> int32 indices, float degree histogram via atomics
// ---------------------------------------------------------------------------
__global__ __launch_bounds__(256) void wss_edge_prep(
    const long long* __restrict__ ei, int* __restrict__ row32,
    int* __restrict__ col32, float* __restrict__ deg, int ne)
{
    int e = blockIdx.x * 256 + threadIdx.x;
    if (e < ne) {
        int r = (int)ei[e];
        int c = (int)ei[(size_t)ne + e];
        row32[e] = r;
        col32[e] = c;
        atomicAdd(&deg[c], 1.0f);
    }
}

__global__ __launch_bounds__(256) void wss_dinv(
    const float* __restrict__ deg, float* __restrict__ dinv, int n)
{
    int i = blockIdx.x * 256 + threadIdx.x;
    if (i < n) {
        float d = deg[i];
        dinv[i] = (d > 0.0f) ? rsqrtf(fmaxf(d, 1.0f)) : 0.0f;
    }
}

__global__ __launch_bounds__(256) void wss_norm(
    const int* __restrict__ row32, const int* __restrict__ col32,
    const float* __restrict__ dinv, float* __restrict__ nrm, int ne)
{
    int e = blockIdx.x * 256 + threadIdx.x;
    if (e < ne) nrm[e] = dinv[row32[e]] * dinv[col32[e]];
}

// ---------------------------------------------------------------------------
// Weight prep: Weff = triu(Wp[:, :-2],1) symmetrized + diag(q*rowsumabs + r)
//              Asym = A - A^T            (both 128x128, one block, 128 thr)
// ---------------------------------------------------------------------------
__global__ __launch_bounds__(128) void wss_weights(
    const float* __restrict__ Wp, const float* __restrict__ A,
    float* __restrict__ Weff, float* __restrict__ Asym)
{
    int i = threadIdx.x;  // row 0..127
    float q = Wp[i * 130 + 128];
    float r = Wp[i * 130 + 129];
    float sum = 0.0f;
    for (int j = 0; j < HID; ++j) {
        float w = (j > i) ? Wp[i * 130 + j] : ((j < i) ? Wp[j * 130 + i] : 0.0f);
        sum += fabsf(w);
    }
    for (int j = 0; j < HID; ++j) {
        float w = (j > i) ? Wp[i * 130 + j] : ((j < i) ? Wp[j * 130 + i] : 0.0f);
        if (j == i) w = q * sum + r;       // S diag is 0
        Weff[i * HID + j] = w;
        Asym[i * HID + j] = A[i * HID + j] - A[j * HID + i];
    }
}

// ---------------------------------------------------------------------------
// GEMM helpers: block = 8 waves, 32 rows x 128 cols per block.
// A operand staged in LDS (padded); B operand pre-held in 32 float2 regs/lane.
// f32 WMMA 16x16x4, K=128 in 32 steps.
// ---------------------------------------------------------------------------
__device__ __forceinline__ void wss_load_tile(
    const float* __restrict__ src, float* sh, int r0, int n)
{
    for (int i = threadIdx.x; i < 32 * (HID / 4); i += 256) {
        int rr = i >> 5, cc = i & 31;
        int row = r0 + rr; if (row >= n) row = n - 1;   // clamp (safe pad)
        *(float4*)&sh[rr * PAD + cc * 4] =
            *(const float4*)(src + (size_t)row * HID + cc * 4);
    }
}

// h = relu(x @ Wi^T + bi)
__global__ __launch_bounds__(256) void wss_gemm_in(
    const float* __restrict__ x, const float* __restrict__ Wi,
    const float* __restrict__ bi, float* __restrict__ h, int n)
{
    __shared__ float sh[32 * PAD];
    const int r0 = blockIdx.x * 32;
    wss_load_tile(x, sh, r0, n);
    __syncthreads();

    const int lane = threadIdx.x & 31;
    const int wid  = threadIdx.x >> 5;
    const int lh   = lane & 15;       // row-in-tile / col-in-tile
    const int kh   = lane >> 4;       // K half selector (0: K0/1, 1: K2/3)
    const int ncol = wid * 16;

    v2f b[32];
    const float* wrow = Wi + (size_t)(ncol + lh) * HID + 2 * kh;
    #pragma unroll
    for (int kk = 0; kk < 32; ++kk) b[kk] = *(const v2f*)(wrow + 4 * kk);

    v8f acc0 = {}; v8f acc1 = {};
    #pragma unroll
    for (int kk = 0; kk < 32; ++kk) {
        v2f a0 = *(const v2f*)&sh[lh * PAD + 4 * kk + 2 * kh];
        v2f a1 = *(const v2f*)&sh[(16 + lh) * PAD + 4 * kk + 2 * kh];
        acc0 = __builtin_amdgcn_wmma_f32_16x16x4_f32(false, a0, false, b[kk],
                                                     (short)0, acc0, false, false);
        acc1 = __builtin_amdgcn_wmma_f32_16x16x4_f32(false, a1, false, b[kk],
                                                     (short)0, acc1, false, false);
    }

    const float bias = bi[ncol + lh];
    #pragma unroll
    for (int r = 0; r < 8; ++r) {
        int row0 = r0 + kh * 8 + r;
        int row1 = row0 + 16;
        if (row0 < n) h[(size_t)row0 * HID + ncol + lh] = fmaxf(acc0[r] + bias, 0.0f);
        if (row1 < n) h[(size_t)row1 * HID + ncol + lh] = fmaxf(acc1[r] + bias, 0.0f);
    }
}

// out = h @ Weff^T ; z = relu(h @ Asym^T)
__global__ __launch_bounds__(256) void wss_gemm_dual(
    const float* __restrict__ h, const float* __restrict__ Weff,
    const float* __restrict__ Asym, float* __restrict__ out,
    float* __restrict__ z, int n)
{
    __shared__ float sh[32 * PAD];
    const int r0 = blockIdx.x * 32;
    wss_load_tile(h, sh, r0, n);
    __syncthreads();

    const int lane = threadIdx.x & 31;
    const int wid  = threadIdx.x >> 5;
    const int lh   = lane & 15;
    const int kh   = lane >> 4;
    const int ncol = wid * 16;

    v2f b1[32], b2[32];
    const float* w1 = Weff + (size_t)(ncol + lh) * HID + 2 * kh;
    const float* w2 = Asym + (size_t)(ncol + lh) * HID + 2 * kh;
    #pragma unroll
    for (int kk = 0; kk < 32; ++kk) { b1[kk] = *(const v2f*)(w1 + 4 * kk);
                                      b2[kk] = *(const v2f*)(w2 + 4 * kk); }

    v8f acc0 = {}, acc1 = {}, acc2 = {}, acc3 = {};
    #pragma unroll
    for (int kk = 0; kk < 32; ++kk) {
        v2f a0 = *(const v2f*)&sh[lh * PAD + 4 * kk + 2 * kh];
        v2f a1 = *(const v2f*)&sh[(16 + lh) * PAD + 4 * kk + 2 * kh];
        acc0 = __builtin_amdgcn_wmma_f32_16x16x4_f32(false, a0, false, b1[kk],
                                                     (short)0, acc0, false, false);
        acc1 = __builtin_amdgcn_wmma_f32_16x16x4_f32(false, a1, false, b1[kk],
                                                     (short)0, acc1, false, false);
        acc2 = __builtin_amdgcn_wmma_f32_16x16x4_f32(false, a0, false, b2[kk],
                                                     (short)0, acc2, false, false);
        acc3 = __builtin_amdgcn_wmma_f32_16x16x4_f32(false, a1, false, b2[kk],
                                                     (short)0, acc3, false, false);
    }

    #pragma unroll
    for (int r = 0; r < 8; ++r) {
        int row0 = r0 + kh * 8 + r;
        int row1 = row0 + 16;
        if (row0 < n) {
            out[(size_t)row0 * HID + ncol + lh] = acc0[r];
            z  [(size_t)row0 * HID + ncol + lh] = fmaxf(acc2[r], 0.0f);
        }
        if (row1 < n) {
            out[(size_t)row1 * HID + ncol + lh] = acc1[r];
            z  [(size_t)row1 * HID + ncol + lh] = fmaxf(acc3[r], 0.0f);
        }
    }
}

// ---------------------------------------------------------------------------
// Edge scatter: one wave per edge, one float4 of features per lane
// ---------------------------------------------------------------------------
__global__ __launch_bounds__(256) void wss_scatter(
    const int* __restrict__ row32, const int* __restrict__ col32,
    const float* __restrict__ nrm, const float* __restrict__ out,
    float* __restrict__ agg, int ne)
{
    unsigned t = blockIdx.x * 256u + threadIdx.x;
    int e = (int)(t >> 5);
    int lane = (int)(t & 31u);
    if (e < ne) {
        int r = row32[e], c = col32[e];
        float nm = nrm[e];
        float4 v = *(const float4*)(out + (size_t)r * HID + lane * 4);
        float* dst = agg + (size_t)c * HID + lane * 4;
        atomicAdd(dst + 0, nm * v.x);
        atomicAdd(dst + 1, nm * v.y);
        atomicAdd(dst + 2, nm * v.z);
        atomicAdd(dst + 3, nm * v.w);
    }
}

// h_out = h + relu(tanh(agg - z))   (float4-vectorized)
__global__ __launch_bounds__(256) void wss_update(
    const float* __restrict__ h, const float* __restrict__ agg,
    const float* __restrict__ z, float* __restrict__ dst, int n4)
{
    int i = blockIdx.x * 256 + threadIdx.x;
    if (i < n4) {
        float4 hv = ((const float4*)h)[i];
        float4 av = ((const float4*)agg)[i];
        float4 zv = ((const float4*)z)[i];
        float4 o;
        o.x = hv.x + fmaxf(tanhf(av.x - zv.x), 0.0f);
        o.y = hv.y + fmaxf(tanhf(av.y - zv.y), 0.0f);
        o.z = hv.z + fmaxf(tanhf(av.z - zv.z), 0.0f);
        o.w = hv.w + fmaxf(tanhf(av.w - zv.w), 0.0f);
        ((float4*)dst)[i] = o;
    }
}

// ---------------------------------------------------------------------------
extern "C" void kernel_launch(void* const* d_in, const int* in_sizes, int n_in,
                              void* d_out, int out_size, void* d_ws, size_t ws_size,
                              hipStream_t stream)
{
    const float*     x  = (const float*)d_in[0];
    const long long* ei = (const long long*)d_in[1];
    const float*     Wi = (const float*)d_in[2];
    const float*     bi = (const float*)d_in[3];
    const float*     A  = (const float*)d_in[4];
    const float*     Wp = (const float*)d_in[5];
    const int n  = in_sizes[0] / HID;   // 100000
    const int ne = in_sizes[1] / 2;     // 1600000
    const int num_layers = 4;           // fixed by reference; can't sync-read in capture

    char* p = (char*)d_ws;
    auto carve = [&](size_t bytes) -> void* {
        void* q = (void*)p; p += (bytes + 255) & ~(size_t)255; return q;
    };
    float* h    = (float*)carve((size_t)n * HID * 4);
    float* ob   = (float*)carve((size_t)n * HID * 4);
    float* zb   = (float*)carve((size_t)n * HID * 4);
    float* agg  = (float*)carve((size_t)n * HID * 4);
    float* deg  = (float*)carve((size_t)n * 4);
    float* dnv  = (float*)carve((size_t)n * 4);
    float* nrm  = (float*)carve((size_t)ne * 4);
    int*   r32  = (int*)  carve((size_t)ne * 4);
    int*   c32  = (int*)  carve((size_t)ne * 4);
    float* Weff = (float*)carve((size_t)HID * HID * 4);
    float* Asym = (float*)carve((size_t)HID * HID * 4);

    const int nb   = (n + 31) / 32;                 // GEMM blocks
    const int ebk  = (ne + 255) / 256;              // per-edge blocks
    const int sbk  = (int)(((unsigned)ne * 32u + 255u) / 256u);
    const int n4   = n * HID / 4;
    const int ubk  = (n4 + 255) / 256;

    hipMemsetAsync(deg, 0, (size_t)n * 4, stream);
    wss_edge_prep<<<ebk, 256, 0, stream>>>(ei, r32, c32, deg, ne);
    wss_dinv<<<(n + 255) / 256, 256, 0, stream>>>(deg, dnv, n);
    wss_norm<<<ebk, 256, 0, stream>>>(r32, c32, dnv, nrm, ne);
    wss_weights<<<1, 128, 0, stream>>>(Wp, A, Weff, Asym);
    wss_gemm_in<<<nb, 256, 0, stream>>>(x, Wi, bi, h, n);

    for (int layer = 0; layer < num_layers; ++layer) {
        hipMemsetAsync(agg, 0, (size_t)n * HID * 4, stream);
        wss_gemm_dual<<<nb, 256, 0, stream>>>(h, Weff, Asym, ob, zb, n);
        wss_scatter<<<sbk, 256, 0, stream>>>(r32, c32, nrm, ob, agg, ne);
        float* dst = (layer == num_layers - 1) ? (float*)d_out : h;
        wss_update<<<ubk, 256, 0, stream>>>(h, agg, zb, dst, n4);
    }
}